// Net_58033598104021
// MI455X (gfx1250) — compile-verified
//
#include <hip/hip_runtime.h>
#include <cmath>

// ---------------- problem constants ----------------
constexpr int KN_FEAT   = 500;
constexpr int KHID      = 128;
constexpr int KCLASSES  = 47;
constexpr int KLAYERS   = 8;

typedef float v2f __attribute__((ext_vector_type(2)));
typedef float v4f __attribute__((ext_vector_type(4)));
typedef float v8f __attribute__((ext_vector_type(8)));
typedef unsigned int u32x4 __attribute__((ext_vector_type(4)));
typedef int i32x8 __attribute__((ext_vector_type(8)));
typedef int i32x4 __attribute__((ext_vector_type(4)));

#if __has_builtin(__builtin_amdgcn_tensor_load_to_lds) && __has_builtin(__builtin_amdgcn_s_wait_tensorcnt)
#define USE_TDM 1
#else
#define USE_TDM 0
#endif

// ---------------- Tensor Data Mover: 1-D global -> LDS copy ----------------
// D# per CDNA5 ISA ch.8: group0 = {count/type/lds_addr/global_addr},
// group1 = {data_size, tensor_dim0, tile_dim0, stride}. 2-D/higher unused.
#if USE_TDM
__device__ inline void tdm_load_1d(unsigned lds_byte_off, const void* gsrc, unsigned ndwords) {
  unsigned long long ga = (unsigned long long)(size_t)gsrc;
  u32x4 g0;
  g0[0] = 1u;                                                  // count=1, user desc
  g0[1] = lds_byte_off;                                        // lds_addr (bytes)
  g0[2] = (unsigned)(ga & 0xFFFFFFFFull);                      // global_addr[31:0]
  g0[3] = (unsigned)((ga >> 32) & 0x01FFFFFFull) | (2u << 30); // addr[56:32] | type=2
  i32x8 g1 = {};
  g1[0] = (int)(2u << 16);                    // data_size=2 -> 4 bytes
  g1[1] = (int)((ndwords & 0xFFFFu) << 16);   // tensor_dim0[15:0]  (bits 63:48)
  g1[2] = (int)((ndwords >> 16) & 0xFFFFu);   // tensor_dim0[31:16] (bits 79:64)
  g1[3] = (int)((ndwords & 0xFFFFu) << 16);   // tile_dim0 (bits 127:112), <=65535
  g1[4] = 0;                                  // tile_dim1/2 = 0 (unused, 1-D)
  g1[5] = (int)ndwords;                       // tensor_dim0_stride[31:0]
  g1[6] = 0;
  g1[7] = 0;
  i32x4 zz4 = {};
  i32x8 zz8 = {};
  // 6-arg form (this toolchain): (g0, g1, g2, g3, g4, cpol)
  __builtin_amdgcn_tensor_load_to_lds(g0, g1, zz4, zz4, zz8, 0);
}
#endif

// Stage ndwords (even) of global data into LDS, whole block participates.
__device__ inline void stage_to_lds(float* bsm, const float* gsrc, int ndwords) {
#if USE_TDM
  if (threadIdx.x < 32) {                 // wave 0 issues one TDM op
    tdm_load_1d((unsigned)(unsigned long long)(void*)bsm, gsrc, (unsigned)ndwords);
    __builtin_amdgcn_s_wait_tensorcnt((short)0);
  }
  __syncthreads();
#else
  for (int i = threadIdx.x; i < (ndwords >> 1); i += blockDim.x)
    ((v2f*)bsm)[i] = ((const v2f*)gsrc)[i];
  __syncthreads();
#endif
}

// ---------------- graph preprocessing ----------------
__global__ void init_kernel(int* __restrict__ deg, int* __restrict__ cursor, int n) {
  int i = blockIdx.x * blockDim.x + threadIdx.x;
  if (i < n) { deg[i] = 0; cursor[i] = 0; }
}

__global__ void count_kernel(const int* __restrict__ dst, int* __restrict__ deg, int e) {
  int i = blockIdx.x * blockDim.x + threadIdx.x;
  if (i < e) atomicAdd(&deg[dst[i]], 1);
}

__global__ void dinv_kernel(const int* __restrict__ deg, float* __restrict__ dinv, int n) {
  int i = blockIdx.x * blockDim.x + threadIdx.x;
  if (i < n) dinv[i] = rsqrtf((float)deg[i] + 1.0f);  // +1 = self loop
}

__global__ void scan_partial_kernel(const int* __restrict__ deg, int* __restrict__ out,
                                    int* __restrict__ bsums, int n) {
  __shared__ int sm[256];
  int tid = threadIdx.x;
  int i = blockIdx.x * 256 + tid;
  int v = (i < n) ? deg[i] : 0;
  sm[tid] = v;
  __syncthreads();
  for (int off = 1; off < 256; off <<= 1) {
    int t = (tid >= off) ? sm[tid - off] : 0;
    __syncthreads();
    sm[tid] += t;
    __syncthreads();
  }
  if (i < n) out[i] = sm[tid] - v;
  if (tid == 255) bsums[blockIdx.x] = sm[255];
}

__global__ void scan_bsums_kernel(int* __restrict__ bsums, int nb,
                                  int* __restrict__ row_ptr, int n) {
  int run = 0;
  for (int b = 0; b < nb; ++b) { int t = bsums[b]; bsums[b] = run; run += t; }
  row_ptr[n] = run;
}

__global__ void scan_add_kernel(int* __restrict__ row_ptr, const int* __restrict__ bsums, int n) {
  int i = blockIdx.x * blockDim.x + threadIdx.x;
  if (i < n) row_ptr[i] += bsums[i >> 8];
}

__global__ void fill_kernel(const int* __restrict__ src, const int* __restrict__ dst,
                            const int* __restrict__ row_ptr, int* __restrict__ cursor,
                            const float* __restrict__ dinv,
                            int* __restrict__ col_idx, float* __restrict__ ew, int e) {
  int i = blockIdx.x * blockDim.x + threadIdx.x;
  if (i < e) {
    int s = src[i], d = dst[i];
    int pos = row_ptr[d] + atomicAdd(&cursor[d], 1);
    col_idx[pos] = s;
    ew[pos] = dinv[s] * dinv[d];
  }
}

// ---------------- weight repack into WMMA-fragment order ----------------
// out[((kg*8 + t)*32 + lane)*2 + j] = in[(4*kg + 2*(lane>>4) + j)*128 + t*16 + (lane&15)]
// so the GEMM inner loop reads one contiguous, coalesced v2f per (kstep,tile).
__global__ void repack_kernel(const float* __restrict__ in, float* __restrict__ out, int total) {
  int idx = blockIdx.x * blockDim.x + threadIdx.x;
  if (idx >= total) return;
  int j    = idx & 1;
  int lane = (idx >> 1) & 31;
  int t    = (idx >> 6) & 7;
  int kg   = idx >> 9;
  int k = 4 * kg + 2 * (lane >> 4) + j;
  int nn = t * 16 + (lane & 15);
  out[idx] = in[k * KHID + nn];
}

// ---------------- GEMM0: h = relu(x @ w0 + b0), K = 500, fp32 WMMA ----------------
// 8 waves/block; wave w owns rows [blk*128+16w, +16) x 128 cols.
// Repacked w0 staged chunk-by-chunk (K=100 per chunk) into LDS via TDM.
__global__ __launch_bounds__(256) void gemm0_kernel(const float* __restrict__ x,
                                                    const float* __restrict__ w0p,
                                                    const float* __restrict__ b0,
                                                    float* __restrict__ h,
                                                    float* __restrict__ h0, int n) {
  constexpr int KG_CHUNK = 25;                      // 25 ksteps * 4 = K=100 per chunk
  constexpr int CHUNK_DW = KG_CHUNK * 8 * 64;       // 12800 floats = 51.2 KB
  __shared__ float bsm[CHUNK_DW];

  const int lane = threadIdx.x & 31;
  const int wave = threadIdx.x >> 5;
  const int half = lane >> 4;
  const int mr   = lane & 15;
  const int m0   = blockIdx.x * 128 + wave * 16;

  int rowA = m0 + mr;
  if (rowA > n - 1) rowA = n - 1;                   // clamp: EXEC all-ones for WMMA
  const float* xrow = x + (size_t)rowA * KN_FEAT;

  v8f acc[8] = {};
  for (int c = 0; c < 5; ++c) {
    stage_to_lds(bsm, w0p + (size_t)c * CHUNK_DW, CHUNK_DW);
    for (int kgl = 0; kgl < KG_CHUNK; ++kgl) {
      const int kb = (c * KG_CHUNK + kgl) * 4 + half * 2;
      v2f a; a.x = xrow[kb]; a.y = xrow[kb + 1];
#pragma unroll
      for (int t = 0; t < 8; ++t) {
        v2f b = *(const v2f*)(bsm + ((kgl * 8 + t) * 32 + lane) * 2);
        acc[t] = __builtin_amdgcn_wmma_f32_16x16x4_f32(false, a, false, b,
                                                       (short)0, acc[t], false, false);
      }
    }
    __syncthreads();                                // LDS reuse barrier
  }
#pragma unroll
  for (int t = 0; t < 8; ++t) {
    const int col = t * 16 + mr;
    const float bias = b0[col];
#pragma unroll
    for (int r = 0; r < 8; ++r) {
      const int row = m0 + r + half * 8;
      if (row < n) {
        float v = acc[t][r] + bias;
        v = v > 0.f ? v : 0.f;
        h [(size_t)row * KHID + col] = v;
        h0[(size_t)row * KHID + col] = v;
      }
    }
  }
}

// ---------------- SpMM: z = 0.9*(self + sum_in norm*h[src]) + 0.1*h0 ----------------
__global__ __launch_bounds__(256) void spmm_kernel(const float* __restrict__ h,
                                                   const float* __restrict__ h0,
                                                   const float* __restrict__ dinv,
                                                   const int* __restrict__ row_ptr,
                                                   const int* __restrict__ col_idx,
                                                   const float* __restrict__ ew,
                                                   float* __restrict__ z, int n) {
  const int lane = threadIdx.x & 31;
  const int wave = threadIdx.x >> 5;
  const int row  = blockIdx.x * 8 + wave;
  if (row >= n) return;
  const int f = lane * 4;

  const float di = dinv[row];
  v4f acc = (di * di) * (*(const v4f*)(h + (size_t)row * KHID + f));

  const int e0 = row_ptr[row], e1 = row_ptr[row + 1];
  for (int e = e0; e < e1; ++e) {
    const int s = col_idx[e];
    const float w = ew[e];
    acc += w * (*(const v4f*)(h + (size_t)s * KHID + f));
  }
  v4f r = *(const v4f*)(h0 + (size_t)row * KHID + f);
  *(v4f*)(z + (size_t)row * KHID + f) = 0.9f * acc + 0.1f * r;
}

// ---------------- layer GEMM: h = relu((1-beta)*z + beta*(z @ W)), K = 128 ----------------
// Whole repacked W (64 KB) staged into LDS once per block via one TDM op.
__global__ __launch_bounds__(256) void layer_gemm_kernel(const float* __restrict__ z,
                                                         const float* __restrict__ Wp,
                                                         float* __restrict__ h,
                                                         float omb, float beta, int n) {
  constexpr int W_DW = 32 * 8 * 64;                 // 16384 floats = 64 KB
  __shared__ float bsm[W_DW];

  const int lane = threadIdx.x & 31;
  const int wave = threadIdx.x >> 5;
  const int half = lane >> 4;
  const int mr   = lane & 15;
  const int m0   = blockIdx.x * 128 + wave * 16;

  int rowA = m0 + mr;
  if (rowA > n - 1) rowA = n - 1;
  const float* zrow = z + (size_t)rowA * KHID;

  stage_to_lds(bsm, Wp, W_DW);

  v8f acc[8] = {};
  for (int kg = 0; kg < 32; ++kg) {
    const int kb = kg * 4 + half * 2;
    v2f a; a.x = zrow[kb]; a.y = zrow[kb + 1];
#pragma unroll
    for (int t = 0; t < 8; ++t) {
      v2f b = *(const v2f*)(bsm + ((kg * 8 + t) * 32 + lane) * 2);
      acc[t] = __builtin_amdgcn_wmma_f32_16x16x4_f32(false, a, false, b,
                                                     (short)0, acc[t], false, false);
    }
  }
#pragma unroll
  for (int t = 0; t < 8; ++t) {
    const int col = t * 16 + mr;
#pragma unroll
    for (int r = 0; r < 8; ++r) {
      const int row = m0 + r + half * 8;
      if (row < n) {
        const float zv = z[(size_t)row * KHID + col];
        float v = omb * zv + beta * acc[t][r];
        v = v > 0.f ? v : 0.f;
        h[(size_t)row * KHID + col] = v;
      }
    }
  }
}

// ---------------- head: logits = h @ w1 + b1; log_softmax; one wave per row ----------------
__global__ __launch_bounds__(256) void final_kernel(const float* __restrict__ h,
                                                    const float* __restrict__ w1,
                                                    const float* __restrict__ b1,
                                                    float* __restrict__ out, int n) {
  const int lane = threadIdx.x & 31;
  const int wave = threadIdx.x >> 5;
  const int row  = blockIdx.x * 8 + wave;
  if (row >= n) return;

  v4f hv = *(const v4f*)(h + (size_t)row * KHID + lane * 4);
  const int  c0   = lane;
  const bool has2 = lane < (KCLASSES - 32);
  const int  c1   = has2 ? (lane + 32) : 0;

  float acc0 = b1[c0];
  float acc1 = has2 ? b1[c1] : 0.f;
  for (int ks = 0; ks < 32; ++ks) {
    const float hb0 = __shfl(hv[0], ks, 32);
    const float hb1 = __shfl(hv[1], ks, 32);
    const float hb2 = __shfl(hv[2], ks, 32);
    const float hb3 = __shfl(hv[3], ks, 32);
    const float* wk = w1 + (size_t)(ks * 4) * KCLASSES;
    acc0 += hb0 * wk[c0] + hb1 * wk[KCLASSES + c0]
          + hb2 * wk[2 * KCLASSES + c0] + hb3 * wk[3 * KCLASSES + c0];
    acc1 += hb0 * wk[c1] + hb1 * wk[KCLASSES + c1]
          + hb2 * wk[2 * KCLASSES + c1] + hb3 * wk[3 * KCLASSES + c1];
  }

  const float v1 = has2 ? acc1 : -__builtin_inff();
  float m = fmaxf(acc0, v1);
  for (int o = 16; o > 0; o >>= 1) m = fmaxf(m, __shfl_xor(m, o, 32));
  float s = expf(acc0 - m) + (has2 ? expf(acc1 - m) : 0.f);
  for (int o = 16; o > 0; o >>= 1) s += __shfl_xor(s, o, 32);
  const float lse = logf(s);
  out[(size_t)row * KCLASSES + c0] = acc0 - m - lse;
  if (has2) out[(size_t)row * KCLASSES + c1] = acc1 - m - lse;
}

// ---------------- host orchestration ----------------
extern "C" void kernel_launch(void* const* d_in, const int* in_sizes, int n_in,
                              void* d_out, int out_size, void* d_ws, size_t ws_size,
                              hipStream_t stream) {
  const float* x     = (const float*)d_in[0];
  const int*   edge  = (const int*)  d_in[1];
  const float* w0    = (const float*)d_in[2];
  const float* b0    = (const float*)d_in[3];
  const float* convw = (const float*)d_in[4];
  const float* w1    = (const float*)d_in[5];
  const float* b1    = (const float*)d_in[6];
  float* out = (float*)d_out;

  const int n = in_sizes[0] / KN_FEAT;   // 100000
  const int E = in_sizes[1] / 2;         // 1600000
  const int* src = edge;
  const int* dst = edge + E;

  // workspace carve-out (256B aligned)
  char* base = (char*)d_ws;
  size_t off = 0;
  auto take = [&](size_t bytes) -> char* {
    char* p = base + off;
    off = (off + bytes + 255) & ~(size_t)255;
    return p;
  };
  int*   deg     = (int*)  take((size_t)n * 4);
  int*   cursor  = (int*)  take((size_t)n * 4);
  float* dinv    = (float*)take((size_t)n * 4);
  int*   row_ptr = (int*)  take((size_t)(n + 1) * 4);
  int*   bsums   = (int*)  take(4096);
  int*   col_idx = (int*)  take((size_t)E * 4);
  float* ew      = (float*)take((size_t)E * 4);
  float* w0p     = (float*)take((size_t)(KN_FEAT / 4) * 8 * 64 * 4);        // 256 KB
  float* convp   = (float*)take((size_t)KLAYERS * (KHID / 4) * 8 * 64 * 4); // 512 KB
  float* h       = (float*)take((size_t)n * KHID * 4);
  float* h0      = (float*)take((size_t)n * KHID * 4);
  float* z       = (float*)take((size_t)n * KHID * 4);
  (void)ws_size; (void)n_in; (void)out_size;

  const int tb = 256;
  const int gN = (n + tb - 1) / tb;
  const int gE = (E + tb - 1) / tb;
  const int nb = (n + 255) / 256;

  // 1) degrees, dinv, CSR build (reused by all 8 layers)
  init_kernel <<<gN, tb, 0, stream>>>(deg, cursor, n);
  count_kernel<<<gE, tb, 0, stream>>>(dst, deg, E);
  dinv_kernel <<<gN, tb, 0, stream>>>(deg, dinv, n);
  scan_partial_kernel<<<nb, 256, 0, stream>>>(deg, row_ptr, bsums, n);
  scan_bsums_kernel  <<<1, 1, 0, stream>>>(bsums, nb, row_ptr, n);
  scan_add_kernel    <<<gN, tb, 0, stream>>>(row_ptr, bsums, n);
  fill_kernel<<<gE, tb, 0, stream>>>(src, dst, row_ptr, cursor, dinv, col_idx, ew, E);

  // 2) weight repack into WMMA-fragment order (one-time, tiny)
  const int t0 = (KN_FEAT / 4) * 8 * 64;   // 64000
  const int tl = (KHID / 4) * 8 * 64;      // 16384
  repack_kernel<<<(t0 + tb - 1) / tb, tb, 0, stream>>>(w0, w0p, t0);
  for (int l = 0; l < KLAYERS; ++l)
    repack_kernel<<<(tl + tb - 1) / tb, tb, 0, stream>>>(convw + (size_t)l * KHID * KHID,
                                                         convp + (size_t)l * tl, tl);

  // 3) input projection (WMMA fp32, TDM-staged weights), writes h and h0
  const int gg = (n + 127) / 128;
  gemm0_kernel<<<gg, 256, 0, stream>>>(x, w0p, b0, h, h0, n);

  // 4) GCNII layers
  const int gs = (n + 7) / 8;
  for (int l = 0; l < KLAYERS; ++l) {
    const float beta = logf(0.5f / (float)(l + 1) + 1.0f);
    spmm_kernel<<<gs, 256, 0, stream>>>(h, h0, dinv, row_ptr, col_idx, ew, z, n);
    layer_gemm_kernel<<<gg, 256, 0, stream>>>(z, convp + (size_t)l * tl, h,
                                              1.0f - beta, beta, n);
  }

  // 5) classifier head + log_softmax
  final_kernel<<<gs, 256, 0, stream>>>(h, w1, b1, out, n);
}